// Reg_version_1_40570261078378
// MI455X (gfx1250) — compile-verified
//
#include <hip/hip_runtime.h>
#include <hip/hip_bf16.h>
#include <stdint.h>

#define N     1024
#define BATCH 64
#define KPB   256          // threads per block = diagonals per block
#define WAVE  32
#define DEPTH 8            // async pipeline depth (per-wave LDS ring)

// ---- CDNA5 async global->LDS path (guarded; falls back to direct loads) ----
#if defined(__AMDGCN__) && \
    __has_builtin(__builtin_amdgcn_global_load_async_to_lds_b32) && \
    __has_builtin(__builtin_amdgcn_s_wait_asynccnt)
#define USE_ASYNC 1
typedef __attribute__((address_space(1))) int gas_int;   // prints as "__device__ int *"
typedef __attribute__((address_space(3))) int las_int;   // LDS pointer
static __device__ __forceinline__ gas_int* as_global(const void* p) {
  return (gas_int*)(unsigned long long)(uintptr_t)p;
}
static __device__ __forceinline__ las_int* as_lds(void* p) {
  // generic LDS address: addr[31:0] is the LDS byte offset (ISA 10.2 aperture map)
  return (las_int*)(unsigned)(uintptr_t)p;
}
#else
#define USE_ASYNC 0
#endif

// Kernel 1: per-(split, batch, k) partial sums s1 = sum(x), s2 = sum(x*x)
// over rows i in [i0, i0+rows_per) of diagonal k (elements attn[b, i, i+k]).
__global__ __launch_bounds__(KPB) void diag_partial_kernel(
    const float* __restrict__ attn,
    float* __restrict__ s1o, float* __restrict__ s2o,
    int rows_per)
{
  const int k  = blockIdx.x * KPB + threadIdx.x;     // 0..N-1
  const int b  = blockIdx.y;
  const int sp = blockIdx.z;
  const int i0 = sp * rows_per;
  const int kw = k & ~(WAVE - 1);                    // wave-uniform min k
  int iend = N - kw;                                 // wave-uniform trip bound
  if (iend > i0 + rows_per) iend = i0 + rows_per;
  const int nk = N - k;                              // valid rows for this lane
  const unsigned NN = (unsigned)N * (unsigned)N;

  const float* base = attn + (size_t)b * N * N;
  float s1 = 0.f, s2 = 0.f;

#if USE_ASYNC
  __shared__ float slab[KPB / WAVE][DEPTH][WAVE];    // 8 KB, per-wave private rings
  const int wv = threadIdx.x / WAVE;
  const int ln = threadIdx.x % WAVE;
  if (iend > i0) {
    // Prime the pipeline: DEPTH rows in flight per wave (128 B each, coalesced).
    #pragma unroll
    for (int d = 0; d < DEPTH; ++d) {
      int r = i0 + d; if (r >= iend) r = iend - 1;   // clamped re-issues hit unread slots
      unsigned off = (unsigned)r * (unsigned)(N + 1) + (unsigned)k;
      if (off >= NN) off = 0u;                       // keep inside this batch slab
      __builtin_amdgcn_global_load_async_to_lds_b32(
          as_global(base + off), as_lds(&slab[wv][(i0 + d) & (DEPTH - 1)][ln]),
          0, 0);
    }
    for (int i = i0; i < iend; ++i) {
      __builtin_amdgcn_s_wait_asynccnt(DEPTH - 1);   // oldest in-flight row landed
      float x = slab[wv][i & (DEPTH - 1)][ln];
      asm volatile("" : "+v"(x));                    // force ds_load (+dscnt wait) before slot reuse
      int r = i + DEPTH; if (r >= iend) r = iend - 1;
      unsigned off = (unsigned)r * (unsigned)(N + 1) + (unsigned)k;
      if (off >= NN) off = 0u;
      __builtin_amdgcn_global_load_async_to_lds_b32(
          as_global(base + off), as_lds(&slab[wv][i & (DEPTH - 1)][ln]),
          0, 0);
      x = (i < nk) ? x : 0.f;                        // mask lanes past their diagonal
      s1 += x;
      s2 = __builtin_fmaf(x, x, s2);
    }
  }
#else
  // Fallback: direct coalesced non-temporal loads + explicit prefetch stream.
  unsigned off = (unsigned)i0 * (unsigned)(N + 1) + (unsigned)k;
  for (int i = i0; i < iend; ++i, off += (N + 1)) {
    unsigned o = (off < NN) ? off : 0u;
    float x = __builtin_nontemporal_load(attn + (size_t)b * N * N + o);
    if (((i - i0) & 15) == 0) {
      unsigned po = off + 32u * (N + 1);
      if (po >= NN) po = 0u;
      __builtin_prefetch(attn + (size_t)b * N * N + po, 0, 0);  // global_prefetch_b8
    }
    x = (i < nk) ? x : 0.f;
    s1 += x;
    s2 = __builtin_fmaf(x, x, s2);
  }
#endif

  const size_t widx = ((size_t)sp * BATCH + b) * N + k;
  s1o[widx] = s1;
  s2o[widx] = s2;
}

// Kernel 2: deterministic final reduction. Combines splits, forms
// std(k) * (N-k)/5 for k = 1..N-2, and averages over k then batch.
__global__ __launch_bounds__(256) void diag_reduce_kernel(
    const float* __restrict__ s1o, const float* __restrict__ s2o,
    float* __restrict__ out, int splits)
{
  __shared__ double red[256];
  const int KVALID = N - 2;                          // k = 1..1022
  const int total = BATCH * KVALID;
  double acc = 0.0;
  for (int t = threadIdx.x; t < total; t += 256) {
    const int b = t / KVALID;
    const int k = 1 + t % KVALID;
    float s1 = 0.f, s2 = 0.f;
    for (int sp = 0; sp < splits; ++sp) {
      const size_t widx = ((size_t)sp * BATCH + b) * N + k;
      s1 += s1o[widx];
      s2 += s2o[widx];
    }
    const float n    = (float)(N - k);
    const float mean = s1 / n;
    float var = s2 - n * mean * mean;
    var = var > 0.f ? var : 0.f;
    var = var / (n - 1.0f);
    acc += (double)(sqrtf(var) * (n * 0.2f));
  }
  red[threadIdx.x] = acc;
  __syncthreads();
  for (int s = 128; s > 0; s >>= 1) {
    if (threadIdx.x < s) red[threadIdx.x] += red[threadIdx.x + s];
    __syncthreads();
  }
  if (threadIdx.x == 0)
    out[0] = (float)(red[0] / ((double)BATCH * (double)KVALID));
}

extern "C" void kernel_launch(void* const* d_in, const int* in_sizes, int n_in,
                              void* d_out, int out_size, void* d_ws, size_t ws_size,
                              hipStream_t stream) {
  (void)in_sizes; (void)n_in; (void)out_size;
  const float* attn = (const float*)d_in[0];
  float* out = (float*)d_out;

  // Split the row range to raise memory-level parallelism (16K waves at 8).
  // Each split needs 2 * BATCH * N floats = 512 KB of workspace.
  const size_t per_split = 2ull * BATCH * N * sizeof(float);
  int splits = 1;
  if      (ws_size >= 8 * per_split) splits = 8;
  else if (ws_size >= 4 * per_split) splits = 4;
  else if (ws_size >= 2 * per_split) splits = 2;

  float* s1o = (float*)d_ws;
  float* s2o = s1o + (size_t)splits * BATCH * N;
  const int rows_per = N / splits;

  dim3 grid(N / KPB, BATCH, splits);
  diag_partial_kernel<<<grid, KPB, 0, stream>>>(attn, s1o, s2o, rows_per);
  diag_reduce_kernel<<<1, 256, 0, stream>>>(s1o, s2o, out, splits);
}